// QuantizedKVCache_43370579755202
// MI455X (gfx1250) — compile-verified
//
#include <hip/hip_runtime.h>

// Problem constants (from reference setup_inputs):
//   B=1, S=8192, H=32, D=128, Snew=128
#define S_TOT 8192
#define H_DIM 32
#define D_DIM 128
#define S_NEW 128
#define TENSOR_ELEMS (S_TOT * H_DIM * D_DIM)  // 33,554,432 elems per tensor

typedef __attribute__((ext_vector_type(4))) int   i32x4;
typedef __attribute__((ext_vector_type(4))) float f32x4;

// Exact pointee type of the async-copy builtin (per hipcc diagnostic):
// "__attribute__((__vector_size__(4 * sizeof(int)))) int __device__ *"
typedef int v4i __attribute__((vector_size(16)));
typedef __attribute__((address_space(1))) v4i g_v4i;  // global (AS1)
typedef __attribute__((address_space(3))) v4i l_v4i;  // LDS (AS3)

// ---- CDNA5 async global->LDS copy (ASYNCcnt path) -------------------------
__device__ __forceinline__ void async_load_b128(const void* gsrc, void* ldst) {
#if __has_builtin(__builtin_amdgcn_global_load_async_to_lds_b128)
  __builtin_amdgcn_global_load_async_to_lds_b128((g_v4i*)gsrc, (l_v4i*)ldst,
                                                 0, 0);
#else
  unsigned loff = (unsigned)(unsigned long long)(l_v4i*)ldst;
  asm volatile("global_load_async_to_lds_b128 %0, %1, off"
               :: "v"(loff), "v"(gsrc) : "memory");
#endif
}

__device__ __forceinline__ void wait_async_zero() {
#if __has_builtin(__builtin_amdgcn_s_wait_asynccnt)
  __builtin_amdgcn_s_wait_asynccnt(0);
  asm volatile("" ::: "memory");  // compiler fence: LDS now holds async data
#else
  asm volatile("s_wait_asynccnt 0" ::: "memory");
#endif
}

// wave32 XOR-lane swizzle (ds_swizzle_b32 group-of-32 encoding:
// offset[14:10]=xor_mask, [4:0]=and_mask=0x1f)
#define SWZ_XOR(x, m) \
  __int_as_float(__builtin_amdgcn_ds_swizzle(__float_as_int(x), (((m) << 10) | 0x1f)))

__device__ __forceinline__ int clamp_start(long long s) {
  int v = (int)s;
  if (v < 0) v = 0;
  if (v > S_TOT - S_NEW) v = S_TOT - S_NEW;  // JAX dynamic_update_slice clamp
  return v;
}

// ---------------------------------------------------------------------------
// Kernel 1: quantize the new K/V tokens, dequantize, write into the update
// window of the output. One wave32 per token (128 floats, 4/lane).
// ---------------------------------------------------------------------------
__global__ void kv_quant_window(const long long* __restrict__ input_pos,
                                const float* __restrict__ k_val,
                                const float* __restrict__ v_val,
                                float* __restrict__ out) {
  const int tid  = threadIdx.x;
  const int lane = tid & 31;
  const int w    = blockIdx.x * 8 + (tid >> 5);  // 8192 waves total
  const int tensor = w >> 12;                    // 0 = K, 1 = V
  const int t      = w & 4095;                   // token within tensor
  const float* __restrict__ src = tensor ? v_val : k_val;
  const int base = t * D_DIM;

  float x0 = src[base + lane];
  float x1 = src[base + lane + 32];
  float x2 = src[base + lane + 64];
  float x3 = src[base + lane + 96];

  float mn = fminf(fminf(x0, x1), fminf(x2, x3));
  float mx = fmaxf(fmaxf(x0, x1), fmaxf(x2, x3));
  mn = fminf(mn, SWZ_XOR(mn, 16)); mx = fmaxf(mx, SWZ_XOR(mx, 16));
  mn = fminf(mn, SWZ_XOR(mn,  8)); mx = fmaxf(mx, SWZ_XOR(mx,  8));
  mn = fminf(mn, SWZ_XOR(mn,  4)); mx = fmaxf(mx, SWZ_XOR(mx,  4));
  mn = fminf(mn, SWZ_XOR(mn,  2)); mx = fmaxf(mx, SWZ_XOR(mx,  2));
  mn = fminf(mn, SWZ_XOR(mn,  1)); mx = fmaxf(mx, SWZ_XOR(mx,  1));

  // Reference qparam math (QMIN=-128, QMAX=127, EPS=FLT_EPSILON)
  mn = fminf(mn, 0.0f);
  mx = fmaxf(mx, 0.0f);
  float scale = fmaxf((mx - mn) / 255.0f, 1.1920928955078125e-07f);
  float dmin = mn / scale;
  float dmax = mx / scale;
  float zp = ((-128.0f + dmin) + (127.0f + dmax) > 0.0f) ? (-128.0f - dmin)
                                                         : (127.0f - dmax);
  zp = rintf(fminf(fmaxf(zp, -128.0f), 127.0f));  // clip then round-half-even

  const int start = clamp_start(input_pos[0]);
  const int snew  = t >> 5;       // token row within new slab
  const int h     = t & 31;
  const size_t obase = (size_t)tensor * TENSOR_ELEMS +
                       ((size_t)(start + snew) * H_DIM + h) * (size_t)D_DIM;

  float xs[4] = {x0, x1, x2, x3};
#pragma unroll
  for (int i = 0; i < 4; ++i) {
    float q = fminf(fmaxf(rintf(xs[i] / scale) + zp, -128.0f), 127.0f);
    __builtin_nontemporal_store((q - zp) * scale, &out[obase + lane + 32 * i]);
  }
}

// ---------------------------------------------------------------------------
// Kernel 2: dequantize the existing int8 cache everywhere outside the update
// window. One block per s-slice (H*D = 4096 int8). Cache bytes are staged
// through LDS via CDNA5 async global->LDS b128 copies; output streamed with
// non-temporal float4 stores (268 MB output > 192 MB L2, so don't cache it).
// ---------------------------------------------------------------------------
__global__ void kv_dequant_cache(const long long* __restrict__ input_pos,
                                 const signed char* __restrict__ k_cache,
                                 const signed char* __restrict__ v_cache,
                                 const float* __restrict__ k_sc,
                                 const float* __restrict__ v_sc,
                                 const long long* __restrict__ k_zp,
                                 const long long* __restrict__ v_zp,
                                 float* __restrict__ out) {
  __shared__ i32x4 stage[256];  // 4 KB: 16 B per thread
  const int tid    = threadIdx.x;
  const int tensor = blockIdx.x >> 13;   // 8192 blocks per tensor
  const int s      = blockIdx.x & 8191;

  const int start = clamp_start(input_pos[0]);
  if (s >= start && s < start + S_NEW) return;  // written by kv_quant_window

  const signed char* __restrict__ cache = tensor ? v_cache : k_cache;
  const float* __restrict__ sc = tensor ? v_sc : k_sc;
  const long long* __restrict__ zp = tensor ? v_zp : k_zp;

  const int off = s * (H_DIM * D_DIM) + tid * 16;  // elem offset in tensor

  async_load_b128(cache + off, &stage[tid]);       // global -> LDS, ASYNCcnt++
  const float scale = sc[off >> 7];                // per-token (128 elems)
  const float zpf   = (float)(int)zp[off >> 7];    // zp fits in [-128,127]
  wait_async_zero();                               // s_wait_asynccnt 0
  const i32x4 v = stage[tid];                      // ds_load_b128

  float* __restrict__ obase = out + (size_t)tensor * TENSOR_ELEMS + (size_t)off;
  const int q[4] = {v.x, v.y, v.z, v.w};
#pragma unroll
  for (int i = 0; i < 4; ++i) {
    const int u = q[i];
    f32x4 r;
    r.x = ((float)(signed char)(u)       - zpf) * scale;
    r.y = ((float)(signed char)(u >> 8)  - zpf) * scale;
    r.z = ((float)(signed char)(u >> 16) - zpf) * scale;
    r.w = ((float)(u >> 24)              - zpf) * scale;  // arithmetic shift
    __builtin_nontemporal_store(r, (f32x4*)(obase + 4 * i));
  }
}

// ---------------------------------------------------------------------------
extern "C" void kernel_launch(void* const* d_in, const int* in_sizes, int n_in,
                              void* d_out, int out_size, void* d_ws,
                              size_t ws_size, hipStream_t stream) {
  (void)in_sizes; (void)n_in; (void)out_size; (void)d_ws; (void)ws_size;

  const long long*   input_pos = (const long long*)d_in[0];
  const float*       k_val     = (const float*)d_in[1];
  const float*       v_val     = (const float*)d_in[2];
  const signed char* k_cache   = (const signed char*)d_in[3];
  const signed char* v_cache   = (const signed char*)d_in[4];
  const float*       k_sc      = (const float*)d_in[5];
  const float*       v_sc      = (const float*)d_in[6];
  const long long*   k_zp      = (const long long*)d_in[7];
  const long long*   v_zp      = (const long long*)d_in[8];
  float* out = (float*)d_out;

  // 8192 tokens (K+V), one wave32 per token, 8 waves per block.
  kv_quant_window<<<1024, 256, 0, stream>>>(input_pos, k_val, v_val, out);

  // 2 tensors * 8192 s-slices, one block per slice (4096 int8 -> 16 KB f32).
  kv_dequant_cache<<<16384, 256, 0, stream>>>(input_pos, k_cache, v_cache,
                                              k_sc, v_sc, k_zp, v_zp, out);
}